// Part_Structure_45372034515258
// MI455X (gfx1250) — compile-verified
//
#include <hip/hip_runtime.h>
#include <hip/hip_bf16.h>

// Problem constants (from reference)
#define B_   64
#define C_   768
#define H_   37
#define HW_  1369
#define SEQ_ 1370
#define D_   768
#define NHID_ 512

typedef __attribute__((ext_vector_type(2))) float v2f;
typedef __attribute__((ext_vector_type(8))) float v8f;

// ---------------------------------------------------------------------------
// Kernel 1: mask[n,j] = sum_c attn[n,c,j]   (the only pass over the 269 MB map)
// Coalesced: threads span j (contiguous), loop over c (stride HW_).
// ---------------------------------------------------------------------------
__global__ void mask_kernel(const float* __restrict__ x, float* __restrict__ mask) {
  int j = blockIdx.x * blockDim.x + threadIdx.x;
  int n = blockIdx.y;
  if (j >= HW_) return;
  const float* p = x + (size_t)n * C_ * HW_ + j;
  float a0 = 0.f, a1 = 0.f, a2 = 0.f, a3 = 0.f;
  #pragma unroll 4
  for (int c = 0; c < C_; c += 4) {
    a0 += p[(size_t)(c + 0) * HW_];
    a1 += p[(size_t)(c + 1) * HW_];
    a2 += p[(size_t)(c + 2) * HW_];
    a3 += p[(size_t)(c + 3) * HW_];
  }
  mask[n * HW_ + j] = (a0 + a1) + (a2 + a3);
}

// ---------------------------------------------------------------------------
// Kernel 2: per-batch statistics. One 256-thread block per n.
// Produces scal[n*8 + {0:alpha, 1:beta, 2:P, 3:M, 4:avg_max}]
//   thr = mean(mask); avg = (mask>thr) ? mask/C : 0
//   idx = argmax(avg);  anchor = grid coord of idx
//   alpha = sum avg*dist, beta = sum avg*ang (dist/ang from anchor geometry)
//   P = sum_{avg>0} avg^2 ; M = sum_{avg<0} avg^2
// ---------------------------------------------------------------------------
__global__ void stats_kernel(const float* __restrict__ mask, float* __restrict__ scal) {
  __shared__ float s_avg[HW_];
  __shared__ float s_red[256];
  __shared__ int   s_idx[256];
  const int n = blockIdx.x;
  const int tx = threadIdx.x;
  const float* m = mask + n * HW_;

  // --- threshold = mean(mask) ---
  float acc = 0.f;
  for (int j = tx; j < HW_; j += 256) acc += m[j];
  s_red[tx] = acc; __syncthreads();
  for (int s = 128; s > 0; s >>= 1) {
    if (tx < s) s_red[tx] += s_red[tx + s];
    __syncthreads();
  }
  const float thr = s_red[0] * (1.0f / (float)HW_);
  __syncthreads();

  // --- avg + argmax(first-index ties) ---
  float bv = -1e30f; int bi = 0;
  for (int j = tx; j < HW_; j += 256) {
    float mj = m[j];
    float av = (mj > thr) ? mj * (1.0f / (float)C_) : 0.0f;
    s_avg[j] = av;
    if (av > bv) { bv = av; bi = j; }
  }
  s_red[tx] = bv; s_idx[tx] = bi; __syncthreads();
  for (int s = 128; s > 0; s >>= 1) {
    if (tx < s) {
      float ov = s_red[tx + s]; int oi = s_idx[tx + s];
      if (ov > s_red[tx] || (ov == s_red[tx] && oi < s_idx[tx])) {
        s_red[tx] = ov; s_idx[tx] = oi;
      }
    }
    __syncthreads();
  }
  const int   idx = s_idx[0];
  const float gmx = s_red[0];
  __syncthreads();

  // --- geometry reductions relative to anchor ---
  const int ai = idx / H_, aj = idx % H_;
  float fa = 0.f, fb = 0.f, fP = 0.f, fM = 0.f;
  const float invH = 1.0f / (float)H_;
  const float invPI = 0.31830988618379067154f;
  for (int j = tx; j < HW_; j += 256) {
    float av = s_avg[j];
    int ii = j / H_, jj = j % H_;
    float r0 = (float)(ii - ai) * invH;
    float r1 = (float)(jj - aj) * invH;
    float sq = r0 * r0 + r1 * r1;
    bool nz = sq > 0.0f;
    float dist = nz ? sqrtf(sq) : 0.0f;
    float ang = atan2f(nz ? r1 : 0.0f, nz ? r0 : 1.0f);
    ang = (ang * invPI + 1.0f) * 0.5f;
    float bin = (m[j] > thr) ? 1.0f : 0.0f;
    dist *= bin; ang *= bin;
    fa += av * dist;
    fb += av * ang;
    if (av > 0.0f)      fP += av * av;
    else if (av < 0.0f) fM += av * av;
  }
  // 4 sequential block reductions (reuse s_red)
  float vals[4] = {fa, fb, fP, fM};
  #pragma unroll
  for (int r = 0; r < 4; ++r) {
    s_red[tx] = vals[r]; __syncthreads();
    for (int s = 128; s > 0; s >>= 1) {
      if (tx < s) s_red[tx] += s_red[tx + s];
      __syncthreads();
    }
    if (tx == 0) scal[n * 8 + r] = s_red[0];
    __syncthreads();
  }
  if (tx == 0) scal[n * 8 + 4] = gmx;
}

// ---------------------------------------------------------------------------
// Kernel 3: WMMA GEMM  U[128,512] @ w2[512,768]  with fused epilogue.
//   A row 2n+0 : relu(+s_n),  row 2n+1 : relu(-s_n),  s_n = a*w1[0] + b*w1[1]
//   epilogue: gathered[n,d] = leaky( g_n * (P_n*U[2n,d] - M_n*U[2n+1,d]) )
// One wave per 16x16 tile, K-loop of 4 via V_WMMA_F32_16X16X4_F32.
// A layout (32b 16x4): lanes 0-15 hold M=lane, K={kb+0,kb+1}; lanes 16-31 M=lane-16,
// K={kb+2,kb+3}. B mirrored with N=lane. C/D: VGPR v, half h -> M = v + 8h.
// ---------------------------------------------------------------------------
__global__ void gcn_wmma_kernel(const float* __restrict__ w1, const float* __restrict__ w2,
                                const float* __restrict__ scal, float* __restrict__ gathered) {
  const int nt = blockIdx.x;          // 0..47  (N tile: d = nt*16 ..)
  const int mt = blockIdx.y;          // 0..7   (M tile: row = mt*16 ..)
  const int lane = threadIdx.x;       // 0..31 (wave32)
  const int mloc = lane & 15;
  const int half = lane >> 4;

  const int row = mt * 16 + mloc;     // A-matrix row for this lane
  const int n = row >> 1;
  const float sgn = (row & 1) ? -1.0f : 1.0f;
  const float alpha = scal[n * 8 + 0];
  const float beta  = scal[n * 8 + 1];
  const int dcol = nt * 16 + mloc;    // B/C column for this lane

  v8f c = {};
  for (int kb = 0; kb < NHID_; kb += 4) {
    const int kA = kb + half * 2;
    float s0 = fmaf(alpha, w1[kA],     beta * w1[NHID_ + kA]);
    float s1 = fmaf(alpha, w1[kA + 1], beta * w1[NHID_ + kA + 1]);
    v2f a, b;
    a.x = fmaxf(sgn * s0, 0.0f);
    a.y = fmaxf(sgn * s1, 0.0f);
    b.x = w2[(size_t)kA * D_ + dcol];
    b.y = w2[(size_t)(kA + 1) * D_ + dcol];
    c = __builtin_amdgcn_wmma_f32_16x16x4_f32(
        /*neg_a=*/false, a, /*neg_b=*/false, b,
        /*c_mod=*/(short)0, c, /*reuse_a=*/false, /*reuse_b=*/false);
  }

  // Fused epilogue: rows (2k, 2k+1) of the tile live in VGPRs (2k, 2k+1) of this lane.
  #pragma unroll
  for (int k2 = 0; k2 < 4; ++k2) {
    int nn = mt * 8 + k2 + 4 * half;       // batch index for this row pair
    float up = c[2 * k2];
    float um = c[2 * k2 + 1];
    float t = scal[nn * 8 + 2] * up - scal[nn * 8 + 3] * um;
    float v = scal[nn * 8 + 4] * t;
    gathered[(size_t)nn * D_ + dcol] = (v > 0.0f) ? v : 0.2f * v;
  }
}

// ---------------------------------------------------------------------------
// Kernel 4: out = hidden_states, with gathered added to seq==0 row. float4.
// ---------------------------------------------------------------------------
__global__ void out_kernel(const float* __restrict__ hs, const float* __restrict__ gathered,
                           float* __restrict__ out) {
  size_t i = (size_t)blockIdx.x * blockDim.x + threadIdx.x;  // float4 index
  const size_t total4 = (size_t)B_ * SEQ_ * D_ / 4;
  if (i >= total4) return;
  float4 v = ((const float4*)hs)[i];
  size_t e = i * 4;
  int d = (int)(e % D_);
  size_t r = e / D_;
  int seq = (int)(r % SEQ_);
  if (seq == 0) {
    int n = (int)(r / SEQ_);
    float4 g = ((const float4*)(gathered + (size_t)n * D_))[d >> 2];
    v.x += g.x; v.y += g.y; v.z += g.z; v.w += g.w;
  }
  ((float4*)out)[i] = v;
}

// ---------------------------------------------------------------------------
extern "C" void kernel_launch(void* const* d_in, const int* in_sizes, int n_in,
                              void* d_out, int out_size, void* d_ws, size_t ws_size,
                              hipStream_t stream) {
  const float* hidden = (const float*)d_in[0];   // (64,1370,768)
  // d_in[1] part_inx, d_in[2] part_value, d_in[4] struc_tokens: unused by reference
  const float* attn = (const float*)d_in[3];     // (64,768,37,37)
  const float* w1 = (const float*)d_in[5];       // (2,512)
  const float* w2 = (const float*)d_in[6];       // (512,768)
  float* out = (float*)d_out;

  float* ws = (float*)d_ws;
  float* mask = ws;                          // 64*1369
  float* scal = mask + (size_t)B_ * HW_;     // 64*8
  float* gath = scal + (size_t)B_ * 8;       // 64*768

  dim3 g1((HW_ + 255) / 256, B_);
  mask_kernel<<<g1, 256, 0, stream>>>(attn, mask);

  stats_kernel<<<B_, 256, 0, stream>>>(mask, scal);

  dim3 g3(D_ / 16, (2 * B_) / 16);           // 48 x 8 tiles, 1 wave each
  gcn_wmma_kernel<<<g3, 32, 0, stream>>>(w1, w2, scal, gath);

  size_t total4 = (size_t)B_ * SEQ_ * D_ / 4;
  out_kernel<<<(unsigned)((total4 + 255) / 256), 256, 0, stream>>>(hidden, gath, out);
}